// SAGENodeCLF_31722628448446
// MI455X (gfx1250) — compile-verified
//
#include <hip/hip_runtime.h>
#include <hip/hip_bf16.h>
#include <math.h>

// ---------------------------------------------------------------------------
// GraphSAGE node classification for MI455X (gfx1250, wave32, WMMA).
// GEMMs use V_WMMA_F32_16X16X4_F32 (full fp32, matches fp32 reference).
// Edge aggregation uses one wave32 per edge + global f32 atomics (L2-resident).
// ---------------------------------------------------------------------------

#define N_NODES 100000
#define N_EDGES 1600000
#define NFEAT   128
#define NHID    128
#define NCLASS  40

typedef __attribute__((ext_vector_type(2))) float v2f;
typedef __attribute__((ext_vector_type(8))) float v8f;

// ----------------------------- utility kernels -----------------------------

__global__ __launch_bounds__(256) void fill_f32(float* __restrict__ p, float v, int n) {
    int i = blockIdx.x * 256 + threadIdx.x;
    if (i < n) p[i] = v;
}

__global__ __launch_bounds__(256) void count_edges(const int* __restrict__ dst,
                                                   float* __restrict__ cnt, int nE) {
    int e = blockIdx.x * 256 + threadIdx.x;
    if (e < nE) atomicAdd(cnt + dst[e], 1.0f);
}

__global__ __launch_bounds__(256) void recip_deg(float* __restrict__ cnt, int n) {
    int i = blockIdx.x * 256 + threadIdx.x;
    if (i < n) cnt[i] = 1.0f / fmaxf(cnt[i], 1.0f);
}

// One wave (32 lanes) per edge; each lane moves 4 contiguous features (float4)
// of h[src] into agg[dst] via f32 atomics. 32 lanes * 4 = 128 features.
__global__ __launch_bounds__(256) void scatter_add_edges(const float* __restrict__ h,
                                                         const int* __restrict__ src,
                                                         const int* __restrict__ dst,
                                                         float* __restrict__ agg, int nE) {
    int e    = blockIdx.x * 8 + (threadIdx.x >> 5);
    int lane = threadIdx.x & 31;
    if (e >= nE) return;
    int s = src[e];
    int d = dst[e];
    const float4* hp = (const float4*)(h + (size_t)s * NFEAT);
    float4 v = hp[lane];
    float* ap = agg + (size_t)d * NFEAT + lane * 4;
    atomicAdd(ap + 0, v.x);
    atomicAdd(ap + 1, v.y);
    atomicAdd(ap + 2, v.z);
    atomicAdd(ap + 3, v.w);
}

// Row-wise log-softmax over NCLASS=40, in place.
__global__ __launch_bounds__(256) void log_softmax_rows(float* __restrict__ out, int n) {
    int i = blockIdx.x * 256 + threadIdx.x;
    if (i >= n) return;
    float* row = out + (size_t)i * NCLASS;
    float m = row[0];
    #pragma unroll
    for (int j = 1; j < NCLASS; ++j) m = fmaxf(m, row[j]);
    float s = 0.0f;
    #pragma unroll
    for (int j = 0; j < NCLASS; ++j) s += expf(row[j] - m);
    float ls = logf(s);
    #pragma unroll
    for (int j = 0; j < NCLASS; ++j) row[j] = row[j] - m - ls;
}

// ----------------------------- WMMA f32 GEMM -------------------------------
// Y[M, NOUT] = act( diag(scale1)*A1 @ W1^T (+ A2 @ W2^T) + bias )
// K = NFEAT = 128 fixed. W row-major [NOUT, 128], so B[k][n] = W[n*128 + k].
// Each wave computes a 16-row x NOUT stripe using NT = ceil(NOUT/16) tiles of
// V_WMMA_F32_16X16X4_F32 accumulators. 8 waves/block -> 128 rows per block.
//
// Register layouts (wave32, per ISA 7.12.2):
//   A 16x4 f32 : lane l (0..15): M=l, K = {0,1}; lane 16..31: M=l-16, K={2,3}
//   B 4x16 f32 : lane l (0..15): N=l, K = {0,1}; lane 16..31: N=l-16, K={2,3}
//   C/D 16x16  : VGPR r, lanes 0..15: M=r, N=l; lanes 16..31: M=r+8, N=l-16

template <int NT, int NOUT, bool HAS_A2>
__global__ __launch_bounds__(256) void gemm_wmma_f32(
    const float* __restrict__ A1, const float* __restrict__ scale1,
    const float* __restrict__ W1,
    const float* __restrict__ A2, const float* __restrict__ W2,
    const float* __restrict__ bias, float* __restrict__ Y,
    int M, int relu)
{
    const int lane = threadIdx.x & 31;
    const int wave = threadIdx.x >> 5;
    const int tile = blockIdx.x * 8 + wave;
    const int m0   = tile * 16;
    if (m0 >= M) return;                 // wave-uniform exit: EXEC stays all-1s

    const int ml  = lane & 15;           // N index within tile / M index for A
    const int kh  = (lane >> 4) * 2;     // K sub-offset for this half-wave
    const int rr  = m0 + ml;
    const int row = (rr < M) ? rr : (M - 1);   // clamp, keep EXEC full
    const float sc = scale1 ? scale1[row] : 1.0f;

    const float* a1p = A1 + (size_t)row * NFEAT;
    const float* a2p = HAS_A2 ? (A2 + (size_t)row * NFEAT) : A1;

    v8f zero = {};
    v8f acc[NT];
    #pragma unroll
    for (int t = 0; t < NT; ++t) acc[t] = zero;

    for (int k = 0; k < NFEAT; k += 4) {
        v2f a;
        a.x = a1p[k + kh + 0] * sc;
        a.y = a1p[k + kh + 1] * sc;
        #pragma unroll
        for (int t = 0; t < NT; ++t) {
            const int n = t * 16 + ml;
            v2f b;
            if (NOUT % 16 == 0 || n < NOUT) {
                b.x = W1[(size_t)n * NFEAT + k + kh + 0];
                b.y = W1[(size_t)n * NFEAT + k + kh + 1];
            } else {
                b.x = 0.0f; b.y = 0.0f;
            }
            acc[t] = __builtin_amdgcn_wmma_f32_16x16x4_f32(
                false, a, false, b, (short)0, acc[t], false, false);
        }
        if (HAS_A2) {
            v2f a2;
            a2.x = a2p[k + kh + 0];
            a2.y = a2p[k + kh + 1];
            #pragma unroll
            for (int t = 0; t < NT; ++t) {
                const int n = t * 16 + ml;
                v2f b;
                if (NOUT % 16 == 0 || n < NOUT) {
                    b.x = W2[(size_t)n * NFEAT + k + kh + 0];
                    b.y = W2[(size_t)n * NFEAT + k + kh + 1];
                } else {
                    b.x = 0.0f; b.y = 0.0f;
                }
                acc[t] = __builtin_amdgcn_wmma_f32_16x16x4_f32(
                    false, a2, false, b, (short)0, acc[t], false, false);
            }
        }
    }

    // Store C/D: m = m0 + r + (lane>=16)*8, n = t*16 + (lane&15)
    const int mbase = m0 + (lane >> 4) * 8;
    #pragma unroll
    for (int t = 0; t < NT; ++t) {
        const int n = t * 16 + ml;
        if (!(NOUT % 16 == 0) && n >= NOUT) continue;
        const float bv = bias ? bias[n] : 0.0f;
        #pragma unroll
        for (int r = 0; r < 8; ++r) {
            const int m = mbase + r;
            if (m < M) {
                float v = acc[t][r] + bv;
                if (relu) v = fmaxf(v, 0.0f);
                Y[(size_t)m * NOUT + n] = v;
            }
        }
    }
}

// ------------------------------- launcher ----------------------------------

extern "C" void kernel_launch(void* const* d_in, const int* in_sizes, int n_in,
                              void* d_out, int out_size, void* d_ws, size_t ws_size,
                              hipStream_t stream) {
    const float* x      = (const float*)d_in[0];
    const int*   ei     = (const int*)d_in[1];
    const float* W_pre  = (const float*)d_in[2];
    const float* b_pre  = (const float*)d_in[3];
    const float* Wl0    = (const float*)d_in[4];
    const float* bl0    = (const float*)d_in[5];
    const float* Wr0    = (const float*)d_in[6];
    const float* Wl1    = (const float*)d_in[7];
    const float* bl1    = (const float*)d_in[8];
    const float* Wr1    = (const float*)d_in[9];
    const float* W_post = (const float*)d_in[10];
    const float* b_post = (const float*)d_in[11];

    const int* src = ei;            // edge_index[0]
    const int* dst = ei + N_EDGES;  // edge_index[1]

    // Workspace layout (floats): agg | h0 | h1 | inv_deg  (~154 MB)
    float* ws  = (float*)d_ws;
    float* agg = ws;
    float* h0  = agg + (size_t)N_NODES * NHID;
    float* h1  = h0  + (size_t)N_NODES * NHID;
    float* inv = h1  + (size_t)N_NODES * NHID;
    float* out = (float*)d_out;

    const int nf   = N_NODES * NHID;                       // 12.8M floats
    const int gFe  = (nf + 255) / 256;                     // feature-buffer fill grid
    const int gN   = (N_NODES + 255) / 256;
    const int gE   = (N_EDGES + 255) / 256;
    const int gEw  = (N_EDGES + 7) / 8;                    // 1 wave per edge
    const int rowTiles = (N_NODES + 15) / 16;              // 6250
    const int gG   = (rowTiles + 7) / 8;                   // 8 waves per block

    // Degree: cnt -> 1/max(cnt,1), stored in `inv`
    fill_f32<<<gN, 256, 0, stream>>>(inv, 0.0f, N_NODES);
    count_edges<<<gE, 256, 0, stream>>>(dst, inv, N_EDGES);
    recip_deg<<<gN, 256, 0, stream>>>(inv, N_NODES);

    // h0 = x @ W_pre^T + b_pre
    gemm_wmma_f32<8, NHID, false><<<gG, 256, 0, stream>>>(
        x, nullptr, W_pre, nullptr, nullptr, b_pre, h0, N_NODES, 0);

    // Layer 0: agg = sum_{src->dst} h0[src];  h1 = relu(agg/deg @ Wl0^T + h0 @ Wr0^T + bl0)
    fill_f32<<<gFe, 256, 0, stream>>>(agg, 0.0f, nf);
    scatter_add_edges<<<gEw, 256, 0, stream>>>(h0, src, dst, agg, N_EDGES);
    gemm_wmma_f32<8, NHID, true><<<gG, 256, 0, stream>>>(
        agg, inv, Wl0, h0, Wr0, bl0, h1, N_NODES, 1);

    // Layer 1: h2 = relu(agg/deg @ Wl1^T + h1 @ Wr1^T + bl1)  (h2 reuses h0 buffer)
    fill_f32<<<gFe, 256, 0, stream>>>(agg, 0.0f, nf);
    scatter_add_edges<<<gEw, 256, 0, stream>>>(h1, src, dst, agg, N_EDGES);
    gemm_wmma_f32<8, NHID, true><<<gG, 256, 0, stream>>>(
        agg, inv, Wl1, h1, Wr1, bl1, h0, N_NODES, 1);

    // logits = h2 @ W_post^T + b_post  (NCLASS=40 -> 3 N-tiles, bounds-guarded)
    gemm_wmma_f32<3, NCLASS, false><<<gG, 256, 0, stream>>>(
        h0, nullptr, W_post, nullptr, nullptr, b_post, out, N_NODES, 0);

    // In-place row-wise log-softmax on d_out
    log_softmax_rows<<<gN, 256, 0, stream>>>(out, N_NODES);
}